// BayesEmbedding_833223656453
// MI455X (gfx1250) — compile-verified
//
#include <hip/hip_runtime.h>
#include <hip/hip_bf16.h>
#include <stdint.h>

// ---------------------------------------------------------------------------
// BayesEmbedding for MI455X (gfx1250, wave32)
//   out[0 .. B*S*D)        : gathered reparameterized embeddings (f32)
//   out[B*S*D]             : KL(q||p) scalar (f32)
// Memory-bound op; CDNA5 paths: global_load_async_to_lds_b128 (ASYNCcnt
// double-buffered gather pipeline), s_wait_asynccnt, global_prefetch_b8.
// Transcendentals kept to raw v_exp_f32/v_log_f32 so the KL pass stays on
// the 23.3 TB/s bandwidth roofline instead of going VALU-bound.
// ---------------------------------------------------------------------------

#define D_DIM      512
#define ROW_BYTES  (D_DIM * 4)      // 2048 bytes per table row

#define KL_BLOCKS  1024
#define KL_THREADS 256

#define GT_THREADS 128              // 4 waves per block
#define GT_BLOCKS  1024             // 4096 waves total -> 4 tokens/wave pipeline

// Cheap log1p: safe here because arguments are exp(-x) in (0,1]; when the
// argument underflows to 0, log(1+0)=0 matches log1p exactly.
__device__ __forceinline__ float fast_log1p(float x) {
    return __logf(1.0f + x);        // v_add_f32 + v_log_f32
}

// softplus(x) = max(x,0) + log1p(exp(-|x|))
__device__ __forceinline__ float softplus_f(float x) {
    float t = __expf(-fabsf(x));
    float s = fast_log1p(t);
    return x > 0.0f ? x + s : s;
}

// KL term for one element. Constants:
//  HALF_LOG_2PI = 0.918938533
//  C1 = log(0.25) - HALF_LOG_2PI - log(SIGMA1) = -1.38629436 - 0.91893853 + 1
//  C2 = log(0.75) - HALF_LOG_2PI - log(SIGMA2) = -0.28768207 - 0.91893853 + 7
//  K1 = 0.5/SIGMA1^2 = 0.5*e^2,  K2 = 0.5/SIGMA2^2 = 0.5*e^14
__device__ __forceinline__ float kl_term(float m, float r, float e) {
    const float HALF_LOG_2PI = 0.91893853320467274f;
    const float C1 = -1.30523289432456199f;
    const float C2 =  5.79337939314815762f;
    const float K1 = 3.69452804946532512f;
    const float K2 = 601302.142082798f;

    float sg = softplus_f(r) + 1e-5f;
    float w  = fmaf(e, sg, m);
    // log q(w) = -0.5*log(2pi) - log(sigma) - 0.5*eps^2   ((w-mu)/sigma == eps)
    float lq = fmaf(-0.5f * e, e, -HALF_LOG_2PI) - __logf(sg);
    float w2 = w * w;
    float l1 = fmaf(-K1, w2, C1);
    float l2 = fmaf(-K2, w2, C2);
    float hi = fmaxf(l1, l2);
    float lo = fminf(l1, l2);
    float lp = hi + fast_log1p(__expf(lo - hi));   // logaddexp
    return lq - lp;
}

// ---------------------------------------------------------------------------
// Pass 1: streaming KL partial sums (deterministic tree reduction per block)
// ---------------------------------------------------------------------------
__global__ __launch_bounds__(KL_THREADS)
void kl_partials_kernel(const float4* __restrict__ mu4,
                        const float4* __restrict__ rho4,
                        const float4* __restrict__ eps4,
                        double* __restrict__ partials, int n4)
{
    double acc = 0.0;
    const int stride = gridDim.x * blockDim.x;
    for (int i = blockIdx.x * blockDim.x + threadIdx.x; i < n4; i += stride) {
        int pf = i + 4 * stride;                 // look-ahead prefetch
        if (pf < n4) {
            __builtin_prefetch(&mu4[pf],  0, 0); // -> global_prefetch_b8
            __builtin_prefetch(&rho4[pf], 0, 0);
            __builtin_prefetch(&eps4[pf], 0, 0);
        }
        float4 m = mu4[i];
        float4 r = rho4[i];
        float4 e = eps4[i];
        float s = kl_term(m.x, r.x, e.x);
        s += kl_term(m.y, r.y, e.y);
        s += kl_term(m.z, r.z, e.z);
        s += kl_term(m.w, r.w, e.w);
        acc += (double)s;
    }

    // wave32 reduction
    for (int o = 16; o > 0; o >>= 1)
        acc += __shfl_down(acc, o, 32);

    __shared__ double red[KL_THREADS / 32];
    const int lane = threadIdx.x & 31;
    const int wv   = threadIdx.x >> 5;
    if (lane == 0) red[wv] = acc;
    __syncthreads();
    if (threadIdx.x == 0) {
        double s = 0.0;
        #pragma unroll
        for (int i = 0; i < KL_THREADS / 32; ++i) s += red[i];
        partials[blockIdx.x] = s;
    }
}

__global__ __launch_bounds__(256)
void kl_finalize_kernel(const double* __restrict__ partials,
                        float* __restrict__ out_kl)
{
    __shared__ double red[256];
    double s = 0.0;
    for (int i = threadIdx.x; i < KL_BLOCKS; i += 256) s += partials[i];
    red[threadIdx.x] = s;
    __syncthreads();
    for (int o = 128; o > 0; o >>= 1) {
        if (threadIdx.x < o) red[threadIdx.x] += red[threadIdx.x + o];
        __syncthreads();
    }
    if (threadIdx.x == 0) *out_kl = (float)red[0];
}

// ---------------------------------------------------------------------------
// Pass 2: embedding gather with async-to-LDS double-buffered pipeline.
// Each wave owns 12 KB of LDS (2 bufs x 3 arrays x 2048 B). Per token it
// issues 12 global_load_async_to_lds_b128 (4 chunks x 3 arrays; 32 lanes x
// 16 B = 512 B per op). While computing token t from buffer b it has token
// t+1 in flight into buffer b^1; s_wait_asynccnt 12 retires exactly the older
// token's ops (async loads complete in order). Lane-private LDS layout ->
// no barriers required.
// ---------------------------------------------------------------------------
#define ASYNC_B128(LDS, VOFF, SBASE, IMM)                                  \
    asm volatile("global_load_async_to_lds_b128 %0, %1, %2 offset:" #IMM   \
                 :: "v"(LDS), "v"(VOFF), "s"(SBASE) : "memory")

#define WAIT_ASYNC(N)                                                      \
    asm volatile("s_wait_asynccnt " #N ::: "memory")

__global__ __launch_bounds__(GT_THREADS)
void gather_kernel(const float* __restrict__ mu,
                   const float* __restrict__ rho,
                   const float* __restrict__ eps,
                   const int*  __restrict__ idx,
                   float* __restrict__ out, int nTok)
{
    // 4 waves * 2 buffers * 3 arrays * 512 floats = 12288 floats = 48 KB
    __shared__ float4 smem4[(GT_THREADS / 32) * 2 * 3 * (D_DIM / 4)];
    float* smem = (float*)smem4;

    const int lane = threadIdx.x & 31;
    const int wv   = threadIdx.x >> 5;
    const int gw   = blockIdx.x * (GT_THREADS >> 5) + wv;
    const int nW   = gridDim.x * (GT_THREADS >> 5);

    float* wbase = smem + wv * (2 * 3 * D_DIM);
    // LDS byte address for this lane (flat-pointer low 32 bits == LDS offset)
    const unsigned ldsLane = (unsigned)(uintptr_t)wbase + (unsigned)(lane * 16);

    auto prefetch = [&](int tok, int buf) {
        int row = idx[tok];                                   // wave-uniform
        unsigned voff = (unsigned)row * (unsigned)ROW_BYTES + (unsigned)(lane * 16);
        unsigned l0 = ldsLane + (unsigned)buf * 6144u;        // mu rows
        unsigned l1 = l0 + 2048u;                             // rho rows
        unsigned l2 = l1 + 2048u;                             // eps rows
        // instruction offset adds to BOTH the global and the LDS address,
        // so 4 chunk ops per array share vaddr/lds vgprs.
        ASYNC_B128(l0, voff, mu,  0);
        ASYNC_B128(l0, voff, mu,  512);
        ASYNC_B128(l0, voff, mu,  1024);
        ASYNC_B128(l0, voff, mu,  1536);
        ASYNC_B128(l1, voff, rho, 0);
        ASYNC_B128(l1, voff, rho, 512);
        ASYNC_B128(l1, voff, rho, 1024);
        ASYNC_B128(l1, voff, rho, 1536);
        ASYNC_B128(l2, voff, eps, 0);
        ASYNC_B128(l2, voff, eps, 512);
        ASYNC_B128(l2, voff, eps, 1024);
        ASYNC_B128(l2, voff, eps, 1536);
    };

    auto compute = [&](int tok, int buf) {
        const float* fb = wbase + buf * 1536;                 // 6144 B in floats
        float* orow = out + (size_t)tok * D_DIM;
        #pragma unroll
        for (int c = 0; c < 4; ++c) {
            int fi = c * 128 + lane * 4;
            float4 m = *(const float4*)&fb[fi];               // ds_load_b128
            float4 r = *(const float4*)&fb[512 + fi];
            float4 e = *(const float4*)&fb[1024 + fi];
            float4 w;
            w.x = fmaf(e.x, softplus_f(r.x) + 1e-5f, m.x);
            w.y = fmaf(e.y, softplus_f(r.y) + 1e-5f, m.y);
            w.z = fmaf(e.z, softplus_f(r.z) + 1e-5f, m.z);
            w.w = fmaf(e.w, softplus_f(r.w) + 1e-5f, m.w);
            *(float4*)&orow[fi] = w;                          // coalesced 512B/wave
        }
    };

    int t = gw;
    if (t >= nTok) return;
    prefetch(t, 0);
    int buf = 0;
    for (; t < nTok; t += nW) {
        int tn = t + nW;
        if (tn < nTok) {
            prefetch(tn, buf ^ 1);   // 12 new ops in flight
            WAIT_ASYNC(12);          // retire the 12 ops of token t
        } else {
            WAIT_ASYNC(0);
        }
        compute(t, buf);
        buf ^= 1;
    }
}

// ---------------------------------------------------------------------------
extern "C" void kernel_launch(void* const* d_in, const int* in_sizes, int n_in,
                              void* d_out, int out_size, void* d_ws, size_t ws_size,
                              hipStream_t stream)
{
    const float* mu  = (const float*)d_in[0];
    const float* rho = (const float*)d_in[1];
    const float* eps = (const float*)d_in[2];
    const int*   idx = (const int*)d_in[3];
    float* out = (float*)d_out;

    const int nElem = in_sizes[0];        // V * D
    const int n4    = nElem >> 2;
    const int nTok  = in_sizes[3];        // B * S

    double* partials = (double*)d_ws;     // KL_BLOCKS doubles
    float*  kl_out   = out + (size_t)nTok * D_DIM;

    hipLaunchKernelGGL(kl_partials_kernel, dim3(KL_BLOCKS), dim3(KL_THREADS), 0, stream,
                       (const float4*)mu, (const float4*)rho, (const float4*)eps,
                       partials, n4);
    hipLaunchKernelGGL(kl_finalize_kernel, dim3(1), dim3(256), 0, stream,
                       partials, kl_out);
    hipLaunchKernelGGL(gather_kernel, dim3(GT_BLOCKS), dim3(GT_THREADS), 0, stream,
                       mu, rho, eps, idx, out, nTok);
    (void)n_in; (void)out_size; (void)ws_size;
}